// WaveletNeuralOperator_1975684956692
// MI455X (gfx1250) — compile-verified
//
#include <hip/hip_runtime.h>
#include <hip/hip_bf16.h>
#include <math.h>

// ---------------- sizes ----------------
#define BB 4
#define CIN 3
#define NN 2048
#define HH 128
#define COUT 3
#define HEADS 8
#define DH 16          // HH / HEADS
#define MM (BB*HH)     // 512 independent 1-D signals

typedef __attribute__((ext_vector_type(2))) float v2f;
typedef __attribute__((ext_vector_type(8))) float v8f;

// f32 WMMA, 16x16 tile, K=4 per instruction (full f32 precision path on CDNA5)
#define WMMA_F32(a_, b_, c_) \
  __builtin_amdgcn_wmma_f32_16x16x4_f32(false, (a_), false, (b_), (short)0, (c_), false, false)

__device__ __forceinline__ float gelu_tanh(float z) {
  float z3 = z * z * z;
  return 0.5f * z * (1.0f + tanhf(0.7978845608028654f * (z + 0.044715f * z3)));
}

// ---------------- 0. collapse W2/b2 over the mean'd output dim ----------------
__global__ void reduce_w2_kernel(const float* __restrict__ W2, const float* __restrict__ b2,
                                 float* __restrict__ W2sum, float* __restrict__ b2sum) {
  int lvl = blockIdx.x;          // 3 blocks
  int i   = threadIdx.x;         // 256 threads
  const float* rowp = W2 + (size_t)(lvl * 256 + i) * 128;
  float s = 0.f;
  for (int j = 0; j < 128; ++j) s += rowp[j];
  W2sum[lvl * 256 + i] = s;
  if (i == 0) {
    float t = 0.f;
    for (int j = 0; j < 128; ++j) t += b2[lvl * 128 + j];
    b2sum[lvl] = t;
  }
}

// ---------------- 1. input projection: (B,CIN,N) @ (CIN,H) -> cur (B*H, N) ----------------
__global__ void input_proj_kernel(const float* __restrict__ x, const float* __restrict__ Win,
                                  const float* __restrict__ b_in, float* __restrict__ xp) {
  int tid = blockIdx.x * blockDim.x + threadIdx.x;   // over MM*NN
  int n  = tid % NN;
  int mh = tid / NN;            // b*HH + h
  int h  = mh % HH;
  int b  = mh / HH;
  float acc = b_in[h];
  #pragma unroll
  for (int c = 0; c < CIN; ++c)
    acc += x[((size_t)(b * CIN + c)) * NN + n] * Win[c * HH + h];
  xp[tid] = acc;
}

// ---------------- 2. fused DWT (stride-2, 8-tap, pad 4/3) + collapsed MLP ----------------
__global__ void dwt_mlp_kernel(const float* __restrict__ cur, int Lprev, int L,
                               const float* __restrict__ wlow, const float* __restrict__ whigh,
                               const float* __restrict__ W1l, const float* __restrict__ b1l,
                               const float* __restrict__ W2sl, const float* __restrict__ b2sl,
                               float* __restrict__ a_out, float* __restrict__ p_out) {
  __shared__ float sW1a[256], sW1d[256], sb1[256], sW2[256];
  int t = threadIdx.x;
  sW1a[t] = W1l[t];          // W1[lvl,0,:]
  sW1d[t] = W1l[256 + t];    // W1[lvl,1,:]
  sb1[t]  = b1l[t];
  sW2[t]  = W2sl[t];
  __syncthreads();

  int tid = blockIdx.x * blockDim.x + threadIdx.x;   // over MM*L
  int m = tid / L;
  int i = tid % L;
  const float* row = cur + (size_t)m * Lprev;

  // a[i] = sum_k x[2i+k-4] * wlow[7-k] ; d[i] idem with whigh
  float av = 0.f, dv = 0.f;
  #pragma unroll
  for (int kk = 0; kk < 8; ++kk) {
    int idx = 2 * i + kk - 4;
    if (idx >= 0 && idx < Lprev) {
      float xv = row[idx];
      av += xv * wlow[7 - kk];
      dv += xv * whigh[7 - kk];
    }
  }
  a_out[tid] = av;

  // p = (gelu(a*W1a + d*W1d + b1) . W2sum + b2sum) / 128
  float acc = 0.f;
  #pragma unroll 4
  for (int ii = 0; ii < 256; ++ii) {
    float z = av * sW1a[ii] + dv * sW1d[ii] + sb1[ii];
    acc += gelu_tanh(z) * sW2[ii];
  }
  p_out[tid] = (acc + b2sl[0]) * (1.0f / 128.0f);
}

// ---------------- 3. IDWT: zero-upsample, pad 4/4, 8-tap, 2-in-channel ----------------
__global__ void idwt_kernel(const float* __restrict__ a, const float* __restrict__ d, int L,
                            const float* __restrict__ wlow, const float* __restrict__ whigh,
                            float* __restrict__ out) {
  int tid = blockIdx.x * blockDim.x + threadIdx.x;   // over MM*2L
  int L2 = 2 * L;
  int m = tid / L2;
  int n = tid % L2;
  const float* ar = a + (size_t)m * L;
  const float* dr = d + (size_t)m * L;
  float acc = 0.f;
  #pragma unroll
  for (int kk = 0; kk < 8; ++kk) {
    int idx = n + kk - 4;
    if (idx >= 0 && (idx & 1) == 0) {
      int j = idx >> 1;
      if (j < L) acc += ar[j] * wlow[kk] + dr[j] * whigh[kk];
    }
  }
  out[tid] = acc;
}

// ---------------- 4. transpose y (B,H,N) -> seq (B*N, H) row-major ----------------
__global__ void transpose_kernel(const float* __restrict__ y, float* __restrict__ seq) {
  int tid = blockIdx.x * blockDim.x + threadIdx.x;   // over MM*NN
  int h  = tid % HH;
  int bn = tid / HH;
  int n = bn % NN;
  int b = bn / NN;
  seq[tid] = y[((size_t)(b * HH + h)) * NN + n];
}

// ---------------- 5. WMMA GEMM: (M x 128) @ (128 x 128) + bias, f32 ----------------
// grid = (M/16, 8), block = 32 (one wave per 16x16 output tile, EXEC all-ones)
__global__ void gemm128_bias_kernel(const float* __restrict__ A, const float* __restrict__ W,
                                    const float* __restrict__ bias, float* __restrict__ out) {
  int lane = threadIdx.x;
  int r = lane & 15;        // free index (row for A, col for B/C)
  int hf = lane >> 4;       // half-wave
  int mt = blockIdx.x, nt = blockIdx.y;
  const float* Arow = A + (size_t)(mt * 16 + r) * 128;
  v8f acc = {};
  for (int k0 = 0; k0 < 128; k0 += 4) {
    int kk = k0 + 2 * hf;
    v2f af; af.x = Arow[kk];                      af.y = Arow[kk + 1];
    v2f bf; bf.x = W[(size_t)kk * 128 + nt * 16 + r];
            bf.y = W[(size_t)(kk + 1) * 128 + nt * 16 + r];
    acc = WMMA_F32(af, bf, acc);
  }
  float bv = bias[nt * 16 + r];
  #pragma unroll
  for (int p = 0; p < 8; ++p)
    out[(size_t)(mt * 16 + p + 8 * hf) * 128 + nt * 16 + r] = acc[p] + bv;
}

// ---------------- 6. flash attention, dh=16, one wave per (b, head, 16-query tile) ----------------
// grid = B*HEADS*(N/16) = 4096, block = 32
__global__ void flash_attn_kernel(const float* __restrict__ q, const float* __restrict__ k,
                                  const float* __restrict__ v, float* __restrict__ o) {
  __shared__ float pmat[256];   // 16x16 P tile, C-layout -> A-layout relay
  int lane = threadIdx.x;
  int r  = lane & 15;
  int hf = lane >> 4;
  int qt   = blockIdx.x & 127;
  int bh   = blockIdx.x >> 7;
  int head = bh & 7;
  int b    = bh >> 3;
  const size_t base = (size_t)b * NN * HH + head * DH;

  // preload Q fragment rows (this wave's 16 queries x 16 dh)
  float qreg[8];
  const float* qrow = q + base + (size_t)(qt * 16 + r) * HH;
  #pragma unroll
  for (int t = 0; t < 4; ++t) {
    int kk = 4 * t + 2 * hf;
    qreg[2 * t]     = qrow[kk];
    qreg[2 * t + 1] = qrow[kk + 1];
  }

  v8f oacc = {};
  float mrow[8], lrow[8];
  #pragma unroll
  for (int p = 0; p < 8; ++p) { mrow[p] = -3.0e38f; lrow[p] = 0.f; }

  for (int kt = 0; kt < NN / 16; ++kt) {
    // S = Q @ K^T  (contraction over dh=16 via 4 chained wmma)
    const float* krow = k + base + (size_t)(kt * 16 + r) * HH;
    v8f s = {};
    #pragma unroll
    for (int t = 0; t < 4; ++t) {
      int kk = 4 * t + 2 * hf;
      v2f af; af.x = qreg[2 * t]; af.y = qreg[2 * t + 1];
      v2f bf; bf.x = krow[kk];    bf.y = krow[kk + 1];
      s = WMMA_F32(af, bf, s);
    }
    // online softmax: each C-layout VGPR p holds row (p + 8*hf) across a 16-lane half
    #pragma unroll
    for (int p = 0; p < 8; ++p) {
      float x = s[p] * 0.25f;   // dh^-0.5
      float mx = x;
      mx = fmaxf(mx, __shfl_xor(mx, 1, 32));
      mx = fmaxf(mx, __shfl_xor(mx, 2, 32));
      mx = fmaxf(mx, __shfl_xor(mx, 4, 32));
      mx = fmaxf(mx, __shfl_xor(mx, 8, 32));
      float mnew = fmaxf(mrow[p], mx);
      float pe = __expf(x - mnew);
      float sum = pe;
      sum += __shfl_xor(sum, 1, 32);
      sum += __shfl_xor(sum, 2, 32);
      sum += __shfl_xor(sum, 4, 32);
      sum += __shfl_xor(sum, 8, 32);
      float corr = __expf(mrow[p] - mnew);
      lrow[p] = lrow[p] * corr + sum;
      mrow[p] = mnew;
      oacc[p] *= corr;
      pmat[(p + 8 * hf) * 16 + r] = pe;   // stage P row-major
    }
    __syncthreads();
    // O += P @ V  (contraction over the 16 keys of this tile)
    #pragma unroll
    for (int t = 0; t < 4; ++t) {
      int jj = 4 * t + 2 * hf;
      v2f af; af.x = pmat[r * 16 + jj]; af.y = pmat[r * 16 + jj + 1];
      v2f bf;
      bf.x = v[base + (size_t)(kt * 16 + jj) * HH + r];
      bf.y = v[base + (size_t)(kt * 16 + jj + 1) * HH + r];
      oacc = WMMA_F32(af, bf, oacc);
    }
    __syncthreads();
  }
  #pragma unroll
  for (int p = 0; p < 8; ++p)
    o[base + (size_t)(qt * 16 + p + 8 * hf) * HH + r] = oacc[p] / lrow[p];
}

// ---------------- 7. final: t (B*N,128) @ Wout(128,3) + bout -> out (B,COUT,N) ----------------
__global__ void out_proj_kernel(const float* __restrict__ t, const float* __restrict__ Wout,
                                const float* __restrict__ bout, float* __restrict__ out) {
  int tid = blockIdx.x * blockDim.x + threadIdx.x;   // over B*COUT*N = 24576
  if (tid >= BB * COUT * NN) return;
  int n  = tid % NN;
  int bc = tid / NN;
  int c = bc % COUT;
  int b = bc / COUT;
  const float* trow = t + ((size_t)b * NN + n) * HH;
  float acc = bout[c];
  for (int h = 0; h < HH; ++h) acc += trow[h] * Wout[h * COUT + c];
  out[tid] = acc;
}

// ---------------- launch ----------------
extern "C" void kernel_launch(void* const* d_in, const int* in_sizes, int n_in,
                              void* d_out, int out_size, void* d_ws, size_t ws_size,
                              hipStream_t stream) {
  (void)in_sizes; (void)n_in; (void)out_size; (void)ws_size;
  const float* x    = (const float*)d_in[0];
  const float* wlow = (const float*)d_in[1];
  const float* whigh= (const float*)d_in[2];
  const float* Win  = (const float*)d_in[3];
  const float* b_in = (const float*)d_in[4];
  const float* W1   = (const float*)d_in[5];
  const float* b1   = (const float*)d_in[6];
  const float* W2   = (const float*)d_in[7];
  const float* b2   = (const float*)d_in[8];
  const float* Wq   = (const float*)d_in[9];
  const float* bq   = (const float*)d_in[10];
  const float* Wk   = (const float*)d_in[11];
  const float* bk   = (const float*)d_in[12];
  const float* Wv   = (const float*)d_in[13];
  const float* bv   = (const float*)d_in[14];
  const float* Wo   = (const float*)d_in[15];
  const float* bo   = (const float*)d_in[16];
  const float* Wout = (const float*)d_in[17];
  const float* bout = (const float*)d_in[18];
  float* out = (float*)d_out;
  float* ws  = (float*)d_ws;

  // workspace layout (floats)
  size_t o_xp    = 0;                        // 512*2048
  size_t o_a1    = o_xp    + (size_t)MM * 2048;
  size_t o_p1    = o_a1    + (size_t)MM * 1024;
  size_t o_a2    = o_p1    + (size_t)MM * 1024;
  size_t o_p2    = o_a2    + (size_t)MM * 512;
  size_t o_a3    = o_p2    + (size_t)MM * 512;
  size_t o_p3    = o_a3    + (size_t)MM * 256;
  size_t o_r512  = o_p3    + (size_t)MM * 256;
  size_t o_r1024 = o_r512  + (size_t)MM * 512;
  size_t o_y     = o_r1024 + (size_t)MM * 1024;
  size_t o_seq   = o_y     + (size_t)MM * 2048;
  size_t o_q     = o_seq   + (size_t)BB * NN * HH;
  size_t o_k     = o_q     + (size_t)BB * NN * HH;
  size_t o_v     = o_k     + (size_t)BB * NN * HH;
  size_t o_o     = o_v     + (size_t)BB * NN * HH;
  size_t o_t     = o_o     + (size_t)BB * NN * HH;
  size_t o_w2s   = o_t     + (size_t)BB * NN * HH;
  size_t o_b2s   = o_w2s   + 3 * 256;

  reduce_w2_kernel<<<dim3(3), dim3(256), 0, stream>>>(W2, b2, ws + o_w2s, ws + o_b2s);

  input_proj_kernel<<<dim3((MM * NN) / 256), dim3(256), 0, stream>>>(x, Win, b_in, ws + o_xp);

  // level 0: 2048 -> 1024 ; level 1: 1024 -> 512 ; level 2: 512 -> 256
  dwt_mlp_kernel<<<dim3((MM * 1024) / 256), dim3(256), 0, stream>>>(
      ws + o_xp, 2048, 1024, wlow, whigh,
      W1 + 0 * 512, b1 + 0 * 256, ws + o_w2s + 0 * 256, ws + o_b2s + 0,
      ws + o_a1, ws + o_p1);
  dwt_mlp_kernel<<<dim3((MM * 512) / 256), dim3(256), 0, stream>>>(
      ws + o_a1, 1024, 512, wlow, whigh,
      W1 + 1 * 512, b1 + 1 * 256, ws + o_w2s + 1 * 256, ws + o_b2s + 1,
      ws + o_a2, ws + o_p2);
  dwt_mlp_kernel<<<dim3((MM * 256) / 256), dim3(256), 0, stream>>>(
      ws + o_a2, 512, 256, wlow, whigh,
      W1 + 2 * 512, b1 + 2 * 256, ws + o_w2s + 2 * 256, ws + o_b2s + 2,
      ws + o_a3, ws + o_p3);

  // reconstruction: rec = p3; rec = idwt(rec, p3) -> idwt(.., p2) -> idwt(.., p1)
  idwt_kernel<<<dim3((MM * 512) / 256), dim3(256), 0, stream>>>(
      ws + o_p3, ws + o_p3, 256, wlow, whigh, ws + o_r512);
  idwt_kernel<<<dim3((MM * 1024) / 256), dim3(256), 0, stream>>>(
      ws + o_r512, ws + o_p2, 512, wlow, whigh, ws + o_r1024);
  idwt_kernel<<<dim3((MM * 2048) / 256), dim3(256), 0, stream>>>(
      ws + o_r1024, ws + o_p1, 1024, wlow, whigh, ws + o_y);

  transpose_kernel<<<dim3((MM * NN) / 256), dim3(256), 0, stream>>>(ws + o_y, ws + o_seq);

  dim3 ggrid(BB * NN / 16, HH / 16);   // (512, 8)
  gemm128_bias_kernel<<<ggrid, dim3(32), 0, stream>>>(ws + o_seq, Wq, bq, ws + o_q);
  gemm128_bias_kernel<<<ggrid, dim3(32), 0, stream>>>(ws + o_seq, Wk, bk, ws + o_k);
  gemm128_bias_kernel<<<ggrid, dim3(32), 0, stream>>>(ws + o_seq, Wv, bv, ws + o_v);

  flash_attn_kernel<<<dim3(BB * HEADS * (NN / 16)), dim3(32), 0, stream>>>(
      ws + o_q, ws + o_k, ws + o_v, ws + o_o);

  gemm128_bias_kernel<<<ggrid, dim3(32), 0, stream>>>(ws + o_o, Wo, bo, ws + o_t);

  out_proj_kernel<<<dim3((BB * COUT * NN + 255) / 256), dim3(256), 0, stream>>>(
      ws + o_t, Wout, bout, out);
}